// global_powermean_pooling_44126493999223
// MI455X (gfx1250) — compile-verified
//
#include <hip/hip_runtime.h>
#include <math.h>

// CDNA5 / gfx1250 power-mean pooling (P=2), one workgroup per graph.
// Segment reduction on the matrix pipe: V_WMMA_F32_16X16X4_F32 with an
// all-ones A matrix sums 4 node-rows x 16 feature-cols per issue into a
// persistent f32 16x16 accumulator. Memory-bound: 512 MB streamed once
// (non-temporal) @ 23.3 TB/s ~ 22 us floor. Main loop is branch-free with
// 8 independent loads in flight per wave; ragged tail is predicated.

typedef __attribute__((ext_vector_type(2))) float v2f;
typedef __attribute__((ext_vector_type(8))) float v8f;

#define D_FEAT 128

__device__ __forceinline__ int lower_bound_i32(const int* __restrict__ b, int n, int key) {
    int lo = 0, hi = n;
    while (lo < hi) {
        int mid = (lo + hi) >> 1;
        if (b[mid] < key) lo = mid + 1; else hi = mid;
    }
    return lo;
}

__global__ __launch_bounds__(128)
void powermean_pool_wmma_kernel(const float* __restrict__ x,
                                const int* __restrict__ batch,
                                float* __restrict__ out,
                                int n_nodes) {
    const int g    = blockIdx.x;          // one graph per workgroup
    const int tid  = threadIdx.x;         // 128 threads = 4 wave32s
    const int lane = tid & 31;
    const int wave = tid >> 5;

    // batch is sorted: find [start, end) of this graph's node rows.
    const int start = lower_bound_i32(batch, n_nodes, g);
    const int end   = lower_bound_i32(batch, n_nodes, g + 1);
    const int cnt   = end - start;

    // This wave owns feature columns [wave*32, wave*32+32) as two 16-col tiles.
    const int colN = lane & 15;
    const int col0 = wave * 32 + colN;
    const int col1 = col0 + 16;
    // B-matrix (4x16 f32) lane layout: lanes 0-15 hold K={0,1}, lanes 16-31 K={2,3}.
    const int rofs = (lane >> 4) * 2;

    // A = all-ones 16x4  ->  every row of D = column sums of B.
    v2f a; a.x = 1.0f; a.y = 1.0f;
    v8f c0 = {};   // accumulator, col0 tile
    v8f c1 = {};   // accumulator, col1 tile

    int r = start;

    // ---- Main loop: 8 rows/iter, unconditional NT loads (8 in flight/wave) ----
    for (; r + 8 <= end; r += 8) {
        const size_t base0 = (size_t)(r + rofs) * D_FEAT;       // rows r+rofs, +1
        const size_t base1 = base0 + 4u * D_FEAT;               // rows r+4+rofs, +1

        float v0 = __builtin_nontemporal_load(&x[base0 + col0]);
        float v1 = __builtin_nontemporal_load(&x[base0 + D_FEAT + col0]);
        float v2 = __builtin_nontemporal_load(&x[base0 + col1]);
        float v3 = __builtin_nontemporal_load(&x[base0 + D_FEAT + col1]);
        float w0 = __builtin_nontemporal_load(&x[base1 + col0]);
        float w1 = __builtin_nontemporal_load(&x[base1 + D_FEAT + col0]);
        float w2 = __builtin_nontemporal_load(&x[base1 + col1]);
        float w3 = __builtin_nontemporal_load(&x[base1 + D_FEAT + col1]);

        v2f b0 = {v0 * v0, v1 * v1};    // x^P, P = 2
        v2f b1 = {v2 * v2, v3 * v3};
        v2f b2 = {w0 * w0, w1 * w1};
        v2f b3 = {w2 * w2, w3 * w3};

        c0 = __builtin_amdgcn_wmma_f32_16x16x4_f32(false, a, false, b0, (short)0, c0, false, false);
        c1 = __builtin_amdgcn_wmma_f32_16x16x4_f32(false, a, false, b1, (short)0, c1, false, false);
        c0 = __builtin_amdgcn_wmma_f32_16x16x4_f32(false, a, false, b2, (short)0, c0, false, false);
        c1 = __builtin_amdgcn_wmma_f32_16x16x4_f32(false, a, false, b3, (short)0, c1, false, false);
    }

    // ---- Ragged tail (< 8 rows): predicated loads, at most 2 iterations ----
    for (; r < end; r += 4) {
        const int rA = r + rofs;
        const int rB = rA + 1;
        float v0 = (rA < end) ? x[(size_t)rA * D_FEAT + col0] : 0.0f;
        float v1 = (rB < end) ? x[(size_t)rB * D_FEAT + col0] : 0.0f;
        float v2 = (rA < end) ? x[(size_t)rA * D_FEAT + col1] : 0.0f;
        float v3 = (rB < end) ? x[(size_t)rB * D_FEAT + col1] : 0.0f;
        v2f b0 = {v0 * v0, v1 * v1};
        v2f b1 = {v2 * v2, v3 * v3};
        c0 = __builtin_amdgcn_wmma_f32_16x16x4_f32(false, a, false, b0, (short)0, c0, false, false);
        c1 = __builtin_amdgcn_wmma_f32_16x16x4_f32(false, a, false, b1, (short)0, c1, false, false);
    }

    // mean then ^(1/P): empty segments -> sqrt(0) = 0, matching reference clamp.
    const float inv = 1.0f / (float)(cnt > 0 ? cnt : 1);

    // C/D layout: VGPR0, lanes 0-15 hold row M=0 (N = lane) -> the column sums.
    if (lane < 16) {
        out[(size_t)g * D_FEAT + col0] = sqrtf(c0[0] * inv);
        out[(size_t)g * D_FEAT + col1] = sqrtf(c1[0] * inv);
    }
}

extern "C" void kernel_launch(void* const* d_in, const int* in_sizes, int n_in,
                              void* d_out, int out_size, void* d_ws, size_t ws_size,
                              hipStream_t stream) {
    (void)n_in; (void)d_ws; (void)ws_size;
    const float* x     = (const float*)d_in[0];
    const int*   batch = (const int*)d_in[1];
    float*       out   = (float*)d_out;

    const int n_nodes  = in_sizes[1];          // 1,000,000
    const int n_graphs = out_size / D_FEAT;    // 4096

    powermean_pool_wmma_kernel<<<n_graphs, 128, 0, stream>>>(x, batch, out, n_nodes);
}